// VMamba_Feature_Extractor_57449482551774
// MI455X (gfx1250) — compile-verified
//
#include <hip/hip_runtime.h>
#include <math.h>

typedef float v2f __attribute__((ext_vector_type(2)));
typedef float v8f __attribute__((ext_vector_type(8)));

__device__ __forceinline__ float gelu_f(float x) {
    return 0.5f * x * (1.0f + erff(x * 0.7071067811865476f));
}
__device__ __forceinline__ float softplus_f(float x) {
    return (x > 20.0f) ? x : log1pf(expf(x));
}
__device__ __forceinline__ float silu_f(float x) {
    return x / (1.0f + expf(-x));
}

// ---------------------------------------------------------------------------
// concat(x0,x1) then 1x1 conv 1->3 channels
__global__ void k_conv1to3(const float* __restrict__ x0, const float* __restrict__ x1,
                           const float* __restrict__ w3, float* __restrict__ out, int HW) {
    int idx = blockIdx.x * blockDim.x + threadIdx.x;
    int total = 8 * 3 * HW;
    if (idx >= total) return;
    int p = idx % HW;
    int o = (idx / HW) % 3;
    int b = idx / (3 * HW);
    const float* src = (b < 4) ? (x0 + (size_t)b * HW) : (x1 + (size_t)(b - 4) * HW);
    out[idx] = w3[o] * src[p];
}

// ---------------------------------------------------------------------------
// generic 3x3 conv, pad=1, NCHW / OIHW, optional groups, act: 0 none, 1 silu
__global__ void k_conv3x3(const float* __restrict__ in, const float* __restrict__ w,
                          const float* __restrict__ bias, float* __restrict__ out,
                          int Bn, int Cin, int Hin, int Win, int Cout, int Hout, int Wout,
                          int stride, int groups, int act) {
    long idx = (long)blockIdx.x * blockDim.x + threadIdx.x;
    long total = (long)Bn * Cout * Hout * Wout;
    if (idx >= total) return;
    int wo = idx % Wout;
    int ho = (idx / Wout) % Hout;
    int co = (idx / ((long)Wout * Hout)) % Cout;
    int b  = idx / ((long)Wout * Hout * Cout);
    int cpg = Cin / groups;
    int opg = Cout / groups;
    int grp = co / opg;
    int ci0 = grp * cpg;
    float acc = bias ? bias[co] : 0.0f;
    int hbase = ho * stride - 1, wbase = wo * stride - 1;
    const float* wp = w + (size_t)co * cpg * 9;
    for (int ci = 0; ci < cpg; ++ci) {
        const float* ip = in + ((size_t)b * Cin + ci0 + ci) * Hin * Win;
        const float* wq = wp + ci * 9;
#pragma unroll
        for (int kh = 0; kh < 3; ++kh) {
            int h = hbase + kh;
            if ((unsigned)h >= (unsigned)Hin) continue;
#pragma unroll
            for (int kw = 0; kw < 3; ++kw) {
                int ww = wbase + kw;
                if ((unsigned)ww >= (unsigned)Win) continue;
                acc += wq[kh * 3 + kw] * ip[(size_t)h * Win + ww];
            }
        }
    }
    if (act == 1) acc = silu_f(acc);
    out[idx] = acc;
}

// ---------------------------------------------------------------------------
// channel LayerNorm per pixel (LN over C), act: 0 none, 1 gelu
__global__ void k_ln2d(const float* __restrict__ in, float* __restrict__ out,
                       const float* __restrict__ gw, const float* __restrict__ bw,
                       int C, int HW, int Bn, int act) {
    int idx = blockIdx.x * blockDim.x + threadIdx.x;
    int total = Bn * HW;
    if (idx >= total) return;
    int b = idx / HW;
    int p = idx % HW;
    const float* ip = in + (size_t)b * C * HW + p;
    float s = 0.0f, s2 = 0.0f;
    for (int c = 0; c < C; ++c) {
        float v = ip[(size_t)c * HW];
        s += v; s2 += v * v;
    }
    float mu = s / C;
    float var = s2 / C - mu * mu;
    float inv = rsqrtf(var + 1e-5f);
    float* op = out + (size_t)b * C * HW + p;
    for (int c = 0; c < C; ++c) {
        float v = (ip[(size_t)c * HW] - mu) * inv * gw[c] + bw[c];
        if (act == 1) v = gelu_f(v);
        op[(size_t)c * HW] = v;
    }
}

// ---------------------------------------------------------------------------
// build 4-direction xs (B,4,C,L) from (B,C,H,W)
__global__ void k_build_xs(const float* __restrict__ x, float* __restrict__ xs,
                           int Bn, int C, int H, int W) {
    int L = H * W;
    long idx = (long)blockIdx.x * blockDim.x + threadIdx.x;
    long total = (long)Bn * C * L;
    if (idx >= total) return;
    int l = idx % L;
    int c = (idx / L) % C;
    int b = idx / ((long)L * C);
    float v = x[idx];
    int h = l / W, w = l % W;
    int lt = w * H + h;
    size_t CL = (size_t)C * L;
    size_t base = ((size_t)(b * 4) * C + c) * L;
    xs[base + l] = v;
    xs[base + CL + lt] = v;
    xs[base + 2 * CL + (L - 1 - l)] = v;
    xs[base + 3 * CL + (L - 1 - lt)] = v;
}

// ---------------------------------------------------------------------------
// fp32 WMMA GEMM: O(M,N) = A(M,K) * B(K,N)  (+bias, epilogue, accumulate)
// One 16x64 tile per wave (4 independent accumulators), K-step 4 via
// V_WMMA_F32_16X16X4_F32. Requirements: N % 64 == 0; M/K edges handled by
// clamped pointers + 0/1 multipliers (branchless main loop).
// epi: 0 none, 1 gelu, 2 softplus
__global__ void k_gemm_wmma(const float* __restrict__ A, const float* __restrict__ Bm,
                            float* __restrict__ O, const float* __restrict__ bias,
                            int M, int N, int K, int lda, int ldb, int ldo,
                            long aStride, int aMod, long bStride, long oStride,
                            long biasStride, int biasMod,
                            int tilesM, int tilesN4, int G, int epi, int accum) {
    int wid = (int)(((long)blockIdx.x * blockDim.x + threadIdx.x) >> 5);
    int lane = threadIdx.x & 31;
    int tilesPerG = tilesM * tilesN4;
    if (wid >= G * tilesPerG) return;  // wave-uniform
    int g = wid / tilesPerG;
    int t = wid - g * tilesPerG;
    int mt = t / tilesN4, nt = t - mt * tilesN4;
    const float* Ap = A + (size_t)(g % aMod) * aStride;
    const float* Bp = Bm + (size_t)g * bStride;
    float* Op = O + (size_t)g * oStride;
    int row = lane & 15, hi = lane >> 4;
    int am = mt * 16 + row;                 // logical A row of this lane
    float amask = (am < M) ? 1.0f : 0.0f;   // loop-invariant edge mask
    int amc = (am < M) ? am : (M - 1);      // clamped (always in-bounds)
    const float* Arow = Ap + (size_t)amc * lda;
    int bn = nt * 64 + row;                 // base column; subtiles at +0/16/32/48
    const float* Bcol = Bp + bn;
    v8f acc0 = {0.f, 0.f, 0.f, 0.f, 0.f, 0.f, 0.f, 0.f};
    v8f acc1 = acc0, acc2 = acc0, acc3 = acc0;
    int K4 = K & ~3;
#pragma unroll 2
    for (int k0 = 0; k0 < K4; k0 += 4) {
        int ka = k0 + hi * 2;               // VGPR j holds K = ka + j
        v2f av = *(const v2f*)(Arow + ka);  // 8B-aligned (lda, ka even)
        av.x *= amask; av.y *= amask;
        const float* b0 = Bcol + (size_t)ka * ldb;
        const float* b1 = Bcol + (size_t)(ka + 1) * ldb;
        v2f bv0 = {b0[0],  b1[0]};
        v2f bv1 = {b0[16], b1[16]};
        v2f bv2 = {b0[32], b1[32]};
        v2f bv3 = {b0[48], b1[48]};
        acc0 = __builtin_amdgcn_wmma_f32_16x16x4_f32(false, av, false, bv0, (short)0, acc0, false, false);
        acc1 = __builtin_amdgcn_wmma_f32_16x16x4_f32(false, av, false, bv1, (short)0, acc1, false, false);
        acc2 = __builtin_amdgcn_wmma_f32_16x16x4_f32(false, av, false, bv2, (short)0, acc2, false, false);
        acc3 = __builtin_amdgcn_wmma_f32_16x16x4_f32(false, av, false, bv3, (short)0, acc3, false, false);
    }
    if (K4 < K) {  // K remainder (only dt GEMM with R=6); masked, runs once
        int ka = K4 + hi * 2;
        float m0 = (ka < K) ? amask : 0.0f;
        float m1 = (ka + 1 < K) ? amask : 0.0f;
        int ka0 = (ka < K) ? ka : 0;
        int ka1 = (ka + 1 < K) ? (ka + 1) : 0;
        v2f av;
        av.x = Arow[ka0] * m0;
        av.y = Arow[ka1] * m1;
        const float* b0 = Bcol + (size_t)ka0 * ldb;
        const float* b1 = Bcol + (size_t)ka1 * ldb;
        v2f bv0 = {b0[0],  b1[0]};
        v2f bv1 = {b0[16], b1[16]};
        v2f bv2 = {b0[32], b1[32]};
        v2f bv3 = {b0[48], b1[48]};
        acc0 = __builtin_amdgcn_wmma_f32_16x16x4_f32(false, av, false, bv0, (short)0, acc0, false, false);
        acc1 = __builtin_amdgcn_wmma_f32_16x16x4_f32(false, av, false, bv1, (short)0, acc1, false, false);
        acc2 = __builtin_amdgcn_wmma_f32_16x16x4_f32(false, av, false, bv2, (short)0, acc2, false, false);
        acc3 = __builtin_amdgcn_wmma_f32_16x16x4_f32(false, av, false, bv3, (short)0, acc3, false, false);
    }
    const float* biasp = bias ? (bias + (size_t)(g % biasMod) * biasStride) : nullptr;
#pragma unroll
    for (int j = 0; j < 8; ++j) {
        int mr = mt * 16 + j + hi * 8;  // C/D layout: VGPR j -> M = j + 8*hi
        if (mr < M) {
            float bv = biasp ? biasp[mr] : 0.0f;
            float* op = Op + (size_t)mr * ldo + bn;
            float v0 = acc0[j] + bv, v1 = acc1[j] + bv, v2 = acc2[j] + bv, v3 = acc3[j] + bv;
            if (epi == 1) { v0 = gelu_f(v0); v1 = gelu_f(v1); v2 = gelu_f(v2); v3 = gelu_f(v3); }
            else if (epi == 2) { v0 = softplus_f(v0); v1 = softplus_f(v1); v2 = softplus_f(v2); v3 = softplus_f(v3); }
            if (accum) { v0 += op[0]; v1 += op[16]; v2 += op[32]; v3 += op[48]; }
            op[0] = v0; op[16] = v1; op[32] = v2; op[48] = v3;
        }
    }
}

// ---------------------------------------------------------------------------
// selective scan: one thread per (b,k,c) chain over L; float4-vectorized
__global__ void k_scan(const float* __restrict__ dts, const float* __restrict__ dbl,
                       const float* __restrict__ xs, const float* __restrict__ A_log,
                       const float* __restrict__ Dp, float* __restrict__ y,
                       int Bn, int C, int L, int Rp2) {
    int idx = blockIdx.x * blockDim.x + threadIdx.x;
    int total = Bn * 4 * C;
    if (idx >= total) return;
    int c = idx % C;
    int k = (idx / C) % 4;
    int b = idx / (4 * C);
    size_t chOff = ((size_t)(b * 4 + k) * C + c) * L;
    const float4* dtp = (const float4*)(dts + chOff);
    const float4* up  = (const float4*)(xs + chOff);
    const float4* bsp = (const float4*)(dbl + ((size_t)(b * 4 + k) * Rp2 + (Rp2 - 2)) * L);
    const float4* csp = (const float4*)(dbl + ((size_t)(b * 4 + k) * Rp2 + (Rp2 - 1)) * L);
    float4* yp = (float4*)(y + chOff);
    float Ac = -expf(A_log[k * C + c]);
    float Dv = Dp[k * C + c];
    float hs = 0.0f;
    int L4 = L >> 2;
    for (int q = 0; q < L4; ++q) {
        float4 dt = dtp[q];
        float4 u  = up[q];
        float4 bs = bsp[q];
        float4 cs = csp[q];
        float4 o;
        hs = expf(dt.x * Ac) * hs + dt.x * bs.x * u.x;  o.x = hs * cs.x + Dv * u.x;
        hs = expf(dt.y * Ac) * hs + dt.y * bs.y * u.y;  o.y = hs * cs.y + Dv * u.y;
        hs = expf(dt.z * Ac) * hs + dt.z * bs.z * u.z;  o.z = hs * cs.z + Dv * u.z;
        hs = expf(dt.w * Ac) * hs + dt.w * bs.w * u.w;  o.w = hs * cs.w + Dv * u.w;
        yp[q] = o;
    }
}

// ---------------------------------------------------------------------------
// combine the 4 scan directions back to (B,C,L)
__global__ void k_combine(const float* __restrict__ y, float* __restrict__ out,
                          int Bn, int C, int H, int W) {
    int L = H * W;
    long idx = (long)blockIdx.x * blockDim.x + threadIdx.x;
    long total = (long)Bn * C * L;
    if (idx >= total) return;
    int l = idx % L;
    int c = (idx / L) % C;
    int b = idx / ((long)L * C);
    int h = l / W, w = l % W;
    int lt = w * H + h;
    size_t CL = (size_t)C * L;
    size_t base = ((size_t)(b * 4) * C + c) * L;
    out[idx] = y[base + l] + y[base + CL + lt] +
               y[base + 2 * CL + (L - 1 - l)] + y[base + 3 * CL + (L - 1 - lt)];
}

// ---------------------------------------------------------------------------
// pixel_shuffle(r=2) of (8,96,64,64) into feat0, split batch halves
__global__ void k_pixshuf_store(const float* __restrict__ x, float* __restrict__ out,
                                size_t off0, size_t off1) {
    long idx = (long)blockIdx.x * blockDim.x + threadIdx.x;
    long total = (long)8 * 24 * 128 * 128;
    if (idx >= total) return;
    int ow = idx & 127;
    int oh = (idx >> 7) & 127;
    int c  = (int)((idx >> 14) % 24);
    int b  = (int)(idx / ((long)24 * 128 * 128));
    int h = oh >> 1, r1 = oh & 1, w = ow >> 1, r2 = ow & 1;
    float v = x[(((size_t)b * 96 + (c * 4 + r1 * 2 + r2)) * 64 + h) * 64 + w];
    size_t rel = (((size_t)(b & 3) * 24 + c) * 128 + oh) * 128 + ow;
    out[(b < 4 ? off0 : off1) + rel] = v;
}

// ---------------------------------------------------------------------------
// copy a (8,C,H,W) feature into d_out, split into batch halves
__global__ void k_store_feat(const float* __restrict__ x, float* __restrict__ out,
                             int perB, size_t off0, size_t off1) {
    long idx = (long)blockIdx.x * blockDim.x + threadIdx.x;
    long total = (long)8 * perB;
    if (idx >= total) return;
    int b = (int)(idx / perB);
    int r = (int)(idx - (long)b * perB);
    size_t rel = (size_t)(b & 3) * perB + r;
    out[(b < 4 ? off0 : off1) + rel] = x[idx];
}

// ---------------------------------------------------------------------------
extern "C" void kernel_launch(void* const* d_in, const int* in_sizes, int n_in,
                              void* d_out, int out_size, void* d_ws, size_t ws_size,
                              hipStream_t stream) {
    (void)in_sizes; (void)n_in; (void)out_size;
    const int DIMS[4] = {96, 192, 384, 768};
    const int DEPTHS[4] = {2, 2, 8, 2};
    auto F = [&](int i) -> const float* { return (const float*)d_in[i]; };

    const float* x0 = F(0);
    const float* x1 = F(1);
    // params flattened in JAX tree order (dict keys sorted; 'A' < 'D' < lowercase)
    int pi = 2;
    const float* conv1to3 = F(pi++);
    const float* pe1_b = F(pi++); const float* pe1_w = F(pi++);
    const float* pe2_b = F(pi++); const float* pe2_w = F(pi++);
    const float* pn1_b = F(pi++); const float* pn1_g = F(pi++);
    const float* pn2_b = F(pi++); const float* pn2_g = F(pi++);
    struct Blk {
        const float *A_log, *D, *conv_w, *dt_b, *dt_w, *fc1_b, *fc1_w, *fc2_b, *fc2_w,
                    *in_proj, *n1_b, *n1_g, *n2_b, *n2_g, *on_b, *on_g, *out_proj, *x_proj;
    };
    struct DSp { const float *b, *bn, *g, *w; };
    Blk blk[4][8];
    DSp dsp[3];
    for (int s = 0; s < 4; ++s) {
        for (int k = 0; k < DEPTHS[s]; ++k) {
            Blk& bl = blk[s][k];
            bl.A_log = F(pi++); bl.D = F(pi++); bl.conv_w = F(pi++);
            bl.dt_b = F(pi++);  bl.dt_w = F(pi++);
            bl.fc1_b = F(pi++); bl.fc1_w = F(pi++);
            bl.fc2_b = F(pi++); bl.fc2_w = F(pi++);
            bl.in_proj = F(pi++);
            bl.n1_b = F(pi++); bl.n1_g = F(pi++);
            bl.n2_b = F(pi++); bl.n2_g = F(pi++);
            bl.on_b = F(pi++); bl.on_g = F(pi++);
            bl.out_proj = F(pi++); bl.x_proj = F(pi++);
        }
        if (s < 3) {
            dsp[s].b = F(pi++); dsp[s].bn = F(pi++);
            dsp[s].g = F(pi++); dsp[s].w = F(pi++);
        }
    }

    // workspace slabs
    const size_t ALN = 256;
    auto aln = [&](size_t x) { return (x + ALN - 1) & ~(ALN - 1); };
    size_t BIG = aln((size_t)8 * 4 * 96 * 4096 * 4);  // 50.3 MB (xs/dts/y/MLP hidden)
    size_t MID = aln((size_t)8 * 96 * 4096 * 4);      // 12.6 MB
    size_t DBL = aln((size_t)8 * 4 * 8 * 4096 * 4);   // 4.2 MB
    size_t need = 3 * BIG + 4 * MID + DBL;
    if (ws_size < need) return;
    char* w = (char*)d_ws;
    float* P0 = (float*)w;                       // xs / MLP hidden / stem tmp
    float* P1 = (float*)(w + BIG);               // dts / stem tmp
    float* P2 = (float*)(w + 2 * BIG);           // y   / stem tmp
    float* P3 = (float*)(w + 3 * BIG);           // X (persistent stage state)
    float* P4 = (float*)(w + 3 * BIG + MID);     // T (LN output)
    float* P5 = (float*)(w + 3 * BIG + 2 * MID); // x1 / downsample out
    float* P6 = (float*)(w + 3 * BIG + 3 * MID); // x2 / ysum
    float* P7 = (float*)(w + 3 * BIG + 4 * MID); // dbl
    float* X = P3;
    float* out = (float*)d_out;

    auto nb = [](long n, int t) { return (unsigned)((n + t - 1) / t); };
    auto gemm = [&](const float* A, const float* Bm, float* O, const float* bias,
                    int M, int N, int K, int lda, int ldb, int ldo,
                    long aStride, int aMod, long bStride, long oStride,
                    long biasStride, int biasMod, int G, int epi, int accum) {
        int tM = (M + 15) / 16, tN4 = N / 64;  // N % 64 == 0 for all call sites
        long waves = (long)G * tM * tN4;
        unsigned blks = nb(waves * 32, 256);
        k_gemm_wmma<<<blks, 256, 0, stream>>>(A, Bm, O, bias, M, N, K, lda, ldb, ldo,
                                              aStride, aMod, bStride, oStride,
                                              biasStride, biasMod, tM, tN4, G, epi, accum);
    };

    // ---- stem ----
    {
        int HW = 256 * 256;
        k_conv1to3<<<nb((long)8 * 3 * HW, 256), 256, 0, stream>>>(x0, x1, conv1to3, P0, HW);
        k_conv3x3<<<nb((long)8 * 48 * 128 * 128, 256), 256, 0, stream>>>(
            P0, pe1_w, pe1_b, P1, 8, 3, 256, 256, 48, 128, 128, 2, 1, 0);
        k_ln2d<<<nb((long)8 * 128 * 128, 256), 256, 0, stream>>>(
            P1, P2, pn1_g, pn1_b, 48, 128 * 128, 8, 1);
        k_conv3x3<<<nb((long)8 * 96 * 64 * 64, 256), 256, 0, stream>>>(
            P2, pe2_w, pe2_b, P4, 8, 48, 128, 128, 96, 64, 64, 2, 1, 0);
        k_ln2d<<<nb((long)8 * 64 * 64, 256), 256, 0, stream>>>(
            P4, X, pn2_g, pn2_b, 96, 64 * 64, 8, 0);
    }

    // ---- output offsets (feats split into batch halves) ----
    size_t featPB[5] = {(size_t)24 * 128 * 128, (size_t)96 * 64 * 64, (size_t)192 * 32 * 32,
                        (size_t)384 * 16 * 16, (size_t)768 * 8 * 8};
    size_t off0[5], off1[5], acc = 0;
    for (int i = 0; i < 5; ++i) { off0[i] = acc; acc += 4 * featPB[i]; }
    for (int i = 0; i < 5; ++i) off1[i] = acc + off0[i];

    k_pixshuf_store<<<nb((long)8 * featPB[0], 256), 256, 0, stream>>>(X, out, off0[0], off1[0]);

    // ---- stages ----
    int Hc = 64, Wc = 64;
    for (int s = 0; s < 4; ++s) {
        int dC = DIMS[s];
        int L = Hc * Wc;
        int R = (dC + 15) / 16;
        int Rp2 = R + 2;
        for (int bi = 0; bi < DEPTHS[s]; ++bi) {
            const Blk& bl = blk[s][bi];
            // t = LN(x)
            k_ln2d<<<nb((long)8 * L, 256), 256, 0, stream>>>(X, P4, bl.n1_g, bl.n1_b, dC, L, 8, 0);
            // x1 = in_proj @ t
            gemm(bl.in_proj, P4, P5, nullptr, dC, L, dC, dC, L, L,
                 0, 1, (long)dC * L, (long)dC * L, 0, 1, 8, 0, 0);
            // x2 = silu(dwconv3x3(x1))
            k_conv3x3<<<nb((long)8 * dC * L, 256), 256, 0, stream>>>(
                P5, bl.conv_w, nullptr, P6, 8, dC, Hc, Wc, dC, Hc, Wc, 1, dC, 1);
            // xs = 4-direction stack
            k_build_xs<<<nb((long)8 * dC * L, 256), 256, 0, stream>>>(P6, P0, 8, dC, Hc, Wc);
            // dbl = x_proj @ xs   (per-direction weights)
            gemm(bl.x_proj, P0, P7, nullptr, Rp2, L, dC, dC, L, L,
                 (long)Rp2 * dC, 4, (long)dC * L, (long)Rp2 * L, 0, 1, 32, 0, 0);
            // dts = softplus(dt_w @ dbl[:R] + dt_b)
            gemm(bl.dt_w, P7, P1, bl.dt_b, dC, L, R, R, L, L,
                 (long)dC * R, 4, (long)Rp2 * L, (long)dC * L, (long)dC, 4, 32, 2, 0);
            // selective scan -> y
            k_scan<<<nb((long)8 * 4 * dC, 256), 256, 0, stream>>>(
                P1, P7, P0, bl.A_log, bl.D, P2, 8, dC, L, Rp2);
            // combine 4 directions -> ysum
            k_combine<<<nb((long)8 * dC * L, 256), 256, 0, stream>>>(P2, P6, 8, dC, Hc, Wc);
            // LN(ysum)
            k_ln2d<<<nb((long)8 * L, 256), 256, 0, stream>>>(P6, P4, bl.on_g, bl.on_b, dC, L, 8, 0);
            // x += out_proj @ .
            gemm(bl.out_proj, P4, X, nullptr, dC, L, dC, dC, L, L,
                 0, 1, (long)dC * L, (long)dC * L, 0, 1, 8, 0, 1);
            // MLP: h = gelu(fc1 @ LN(x) + b1); x += fc2 @ h + b2
            k_ln2d<<<nb((long)8 * L, 256), 256, 0, stream>>>(X, P4, bl.n2_g, bl.n2_b, dC, L, 8, 0);
            gemm(bl.fc1_w, P4, P0, bl.fc1_b, 4 * dC, L, dC, dC, L, L,
                 0, 1, (long)dC * L, (long)4 * dC * L, 0, 1, 8, 1, 0);
            gemm(bl.fc2_w, P0, X, bl.fc2_b, dC, L, 4 * dC, 4 * dC, L, L,
                 0, 1, (long)4 * dC * L, (long)dC * L, 0, 1, 8, 0, 1);
        }
        // emit feature map
        k_store_feat<<<nb((long)8 * featPB[s + 1], 256), 256, 0, stream>>>(
            X, out, (int)featPB[s + 1], off0[s + 1], off1[s + 1]);
        // downsample
        if (s < 3) {
            int Ho = Hc / 2, Wo = Wc / 2, Co = 2 * dC;
            k_conv3x3<<<nb((long)8 * Co * Ho * Wo, 256), 256, 0, stream>>>(
                X, dsp[s].w, dsp[s].b, P5, 8, dC, Hc, Wc, Co, Ho, Wo, 2, 1, 0);
            k_ln2d<<<nb((long)8 * Ho * Wo, 256), 256, 0, stream>>>(
                P5, X, dsp[s].g, dsp[s].bn, Co, Ho * Wo, 8, 0);
            Hc = Ho; Wc = Wo;
        }
    }
}